// GCNBackbone_69595650065050
// MI455X (gfx1250) — compile-verified
//
#include <hip/hip_runtime.h>

typedef __attribute__((ext_vector_type(2))) float v2f;
typedef __attribute__((ext_vector_type(8))) float v8f;

#define NNODES 50000
#define DIM 256

// Scheduling pipeline hints: force "5 VMEM loads, then 4 WMMAs" per unrolled
// K-step so prefetches overlap matrix math instead of being sunk to their uses.
#if __has_builtin(__builtin_amdgcn_sched_group_barrier)
#define SCHED_GROUP(mask, size, id) __builtin_amdgcn_sched_group_barrier(mask, size, id)
#else
#define SCHED_GROUP(mask, size, id)
#endif

// ---------------------------------------------------------------------------
// Degree / normalization kernels
// ---------------------------------------------------------------------------
__global__ void init_deg_kernel(float* deg, int n) {
    int i = blockIdx.x * blockDim.x + threadIdx.x;
    if (i < n) deg[i] = 1.0f;  // self-loop contributes 1 to every degree
}

__global__ void count_deg_kernel(const int* __restrict__ dst, float* deg, int E) {
    int e = blockIdx.x * blockDim.x + threadIdx.x;
    if (e < E) atomicAdd(&deg[dst[e]], 1.0f);
}

__global__ void make_dinv_kernel(float* deg, int n) {
    int i = blockIdx.x * blockDim.x + threadIdx.x;
    if (i < n) {
        float d = deg[i];
        deg[i] = (d > 0.0f) ? rsqrtf(d) : 0.0f;
    }
}

// ---------------------------------------------------------------------------
// Repack W[256][256] row-major into K-pair-interleaved layout so the WMMA
// B fragment (elements W[kk][n], W[kk+1][n]) is one contiguous b64 load:
//   Wp[(kp*DIM + n)*2 + j] = W[(2*kp + j)*DIM + n]
// ---------------------------------------------------------------------------
__global__ __launch_bounds__(256) void repack_W_kernel(
    const float* __restrict__ W, float* __restrict__ Wp)
{
    int idx = blockIdx.x * blockDim.x + threadIdx.x;   // 128*256 threads
    int kp = idx >> 8;
    int n  = idx & 255;
    float2 v;
    v.x = W[(size_t)(2 * kp)     * DIM + n];
    v.y = W[(size_t)(2 * kp + 1) * DIM + n];
    ((float2*)Wp)[(size_t)kp * DIM + n] = v;
}

// ---------------------------------------------------------------------------
// GEMM: H[M,256] = X[M,256] @ W[256,256] using V_WMMA_F32_16X16X4_F32.
// Block = 256 threads = 8 waves. Each wave computes a 64(M) x 16(N) tile as
// 4 stacked 16x16 WMMA tiles sharing one B fragment per K-step.
// Software double-buffered; sched_group_barrier pins the issue order to
// (5 loads for k+4) -> (4 WMMAs for k) per unrolled step.
// grid = (ceil(M/64), 2)
// ---------------------------------------------------------------------------
__global__ __launch_bounds__(256) void gemm_wmma_f32(
    const float* __restrict__ X, const float* __restrict__ Wp,
    float* __restrict__ H, int M)
{
    const int lane  = threadIdx.x & 31;
    const int wave  = threadIdx.x >> 5;
    const int m0    = blockIdx.x * 64;
    const int n0    = blockIdx.y * 128 + wave * 16;
    const int lhalf = lane >> 4;   // 0: K pair {k,k+1}, 1: K pair {k+2,k+3}
    const int lmod  = lane & 15;

    v8f c0 = {}, c1 = {}, c2 = {}, c3 = {};

    // Clamp A row indices so EXEC stays all-ones through the WMMAs; the
    // partial-tile garbage is masked at the store.
    int r0 = m0 +  0 + lmod; if (r0 >= M) r0 = M - 1;
    int r1 = m0 + 16 + lmod; if (r1 >= M) r1 = M - 1;
    int r2 = m0 + 32 + lmod; if (r2 >= M) r2 = M - 1;
    int r3 = m0 + 48 + lmod; if (r3 >= M) r3 = M - 1;

    const float* xa0 = X + (size_t)r0 * DIM + 2 * lhalf;
    const float* xa1 = X + (size_t)r1 * DIM + 2 * lhalf;
    const float* xa2 = X + (size_t)r2 * DIM + 2 * lhalf;
    const float* xa3 = X + (size_t)r3 * DIM + 2 * lhalf;
    // B fragment base: one v2f per (kpair, col); kpair for this lane = k/2+lhalf
    const v2f* wb = (const v2f*)Wp + (size_t)lhalf * DIM + n0 + lmod;

    // --- pipeline prologue: fragments for k = 0 ---
    v2f bC  = wb[0];
    v2f a0C = *(const v2f*)(xa0);
    v2f a1C = *(const v2f*)(xa1);
    v2f a2C = *(const v2f*)(xa2);
    v2f a3C = *(const v2f*)(xa3);

    // --- steady state: prefetch k+4, compute k ---  (63 iterations = 9*7)
    #pragma unroll 7
    for (int k = 0; k < DIM - 4; k += 4) {
        v2f bN  = wb[(size_t)((k + 4) >> 1) * DIM];
        v2f a0N = *(const v2f*)(xa0 + k + 4);
        v2f a1N = *(const v2f*)(xa1 + k + 4);
        v2f a2N = *(const v2f*)(xa2 + k + 4);
        v2f a3N = *(const v2f*)(xa3 + k + 4);

        c0 = __builtin_amdgcn_wmma_f32_16x16x4_f32(false, a0C, false, bC, (short)0, c0, false, false);
        c1 = __builtin_amdgcn_wmma_f32_16x16x4_f32(false, a1C, false, bC, (short)0, c1, false, false);
        c2 = __builtin_amdgcn_wmma_f32_16x16x4_f32(false, a2C, false, bC, (short)0, c2, false, false);
        c3 = __builtin_amdgcn_wmma_f32_16x16x4_f32(false, a3C, false, bC, (short)0, c3, false, false);

        bC = bN; a0C = a0N; a1C = a1N; a2C = a2N; a3C = a3N;

        // Pin issue order: all 5 prefetch loads, then the 4 WMMAs.
        SCHED_GROUP(0x0020, 5, 0);  // VMEM read group
        SCHED_GROUP(0x0008, 4, 0);  // MFMA/WMMA group
    }

    // --- epilogue: compute last K-step ---
    c0 = __builtin_amdgcn_wmma_f32_16x16x4_f32(false, a0C, false, bC, (short)0, c0, false, false);
    c1 = __builtin_amdgcn_wmma_f32_16x16x4_f32(false, a1C, false, bC, (short)0, c1, false, false);
    c2 = __builtin_amdgcn_wmma_f32_16x16x4_f32(false, a2C, false, bC, (short)0, c2, false, false);
    c3 = __builtin_amdgcn_wmma_f32_16x16x4_f32(false, a3C, false, bC, (short)0, c3, false, false);

    const int col = n0 + lmod;
    if (m0 + 64 <= M) {
        // Fast path (781 of 782 M-blocks): no bounds checks.
        #pragma unroll
        for (int r = 0; r < 8; ++r) {
            int row = m0 + r + 8 * lhalf;
            H[(size_t)row * DIM + col]        = c0[r];
            H[(size_t)(row + 16) * DIM + col] = c1[r];
            H[(size_t)(row + 32) * DIM + col] = c2[r];
            H[(size_t)(row + 48) * DIM + col] = c3[r];
        }
    } else {
        #pragma unroll
        for (int r = 0; r < 8; ++r) {
            int row = m0 + r + 8 * lhalf;
            if (row < M)      H[(size_t)row * DIM + col]        = c0[r];
            if (row + 16 < M) H[(size_t)(row + 16) * DIM + col] = c1[r];
            if (row + 32 < M) H[(size_t)(row + 32) * DIM + col] = c2[r];
            if (row + 48 < M) H[(size_t)(row + 48) * DIM + col] = c3[r];
        }
    }
}

// ---------------------------------------------------------------------------
// Edge scatter: agg[dst[e]] += H[src[e]] * dinv[src]*dinv[dst]
// 4 edges per 256-thread block; 64 threads/edge; float4 gather, f32 atomics.
// ---------------------------------------------------------------------------
__global__ __launch_bounds__(256) void scatter_edges(
    const float* __restrict__ H, const int* __restrict__ src,
    const int* __restrict__ dst, const float* __restrict__ dinv,
    float* __restrict__ agg, int E)
{
    int e = blockIdx.x * 4 + (threadIdx.x >> 6);
    if (e >= E) return;
    int t = threadIdx.x & 63;  // 64 threads * 4 floats = 256 features

    int s = src[e];
    int d = dst[e];
    float norm = dinv[s] * dinv[d];

    float4 hv = ((const float4*)(H + (size_t)s * DIM))[t];
    float* out = agg + (size_t)d * DIM + t * 4;
    atomicAdd(out + 0, hv.x * norm);
    atomicAdd(out + 1, hv.y * norm);
    atomicAdd(out + 2, hv.z * norm);
    atomicAdd(out + 3, hv.w * norm);
}

// ---------------------------------------------------------------------------
// Finalize (in place): agg = relu(agg + H*dinv^2 (self-loop) + bias)
// One block per node, one thread per feature.
// ---------------------------------------------------------------------------
__global__ __launch_bounds__(256) void finalize_relu(
    float* __restrict__ agg, const float* __restrict__ H,
    const float* __restrict__ dinv, const float* __restrict__ bias)
{
    int node = blockIdx.x;
    int f    = threadIdx.x;
    size_t idx = (size_t)node * DIM + f;
    float di = dinv[node];
    float v = agg[idx] + H[idx] * di * di + bias[f];
    agg[idx] = v > 0.0f ? v : 0.0f;
}

// ---------------------------------------------------------------------------
extern "C" void kernel_launch(void* const* d_in, const int* in_sizes, int n_in,
                              void* d_out, int out_size, void* d_ws, size_t ws_size,
                              hipStream_t stream)
{
    const float* x  = (const float*)d_in[0];
    const int*   ei = (const int*)d_in[1];
    const float* W1 = (const float*)d_in[2];
    const float* b1 = (const float*)d_in[3];
    const float* W2 = (const float*)d_in[4];
    const float* b2 = (const float*)d_in[5];

    const int E = in_sizes[1] / 2;
    const int* src = ei;
    const int* dst = ei + E;

    char*  ws   = (char*)d_ws;
    float* dinv = (float*)ws;                                   // NNODES f32
    size_t off  = ((size_t)NNODES * sizeof(float) + 255) & ~(size_t)255;
    float* Wp   = (float*)(ws + off);                           // 256 KB repacked W
    size_t woff = off + (size_t)DIM * DIM * sizeof(float);
    size_t fsz  = (size_t)NNODES * DIM * sizeof(float);         // 51.2 MB
    float* B0   = (float*)(ws + woff);         // dense GEMM result H
    float* B1   = (float*)(ws + woff + fsz);   // layer-1 aggregate / layer-2 input
    float* OUT  = (float*)d_out;

    // Degrees -> dinv (shared by both layers)
    init_deg_kernel<<<(NNODES + 255) / 256, 256, 0, stream>>>(dinv, NNODES);
    count_deg_kernel<<<(E + 255) / 256, 256, 0, stream>>>(dst, dinv, E);
    make_dinv_kernel<<<(NNODES + 255) / 256, 256, 0, stream>>>(dinv, NNODES);

    dim3 ggrid((NNODES + 63) / 64, 2);

    // Layer 1: B1 = relu(norm_agg(x @ W1) + b1)
    repack_W_kernel<<<(DIM / 2), 256, 0, stream>>>(W1, Wp);
    gemm_wmma_f32<<<ggrid, 256, 0, stream>>>(x, Wp, B0, NNODES);
    hipMemsetAsync(B1, 0, fsz, stream);
    scatter_edges<<<(E + 3) / 4, 256, 0, stream>>>(B0, src, dst, dinv, B1, E);
    finalize_relu<<<NNODES, 256, 0, stream>>>(B1, B0, dinv, b1);

    // Layer 2: OUT = relu(norm_agg(B1 @ W2) + b2)
    repack_W_kernel<<<(DIM / 2), 256, 0, stream>>>(W2, Wp);
    gemm_wmma_f32<<<ggrid, 256, 0, stream>>>(B1, Wp, B0, NNODES);
    hipMemsetAsync(OUT, 0, fsz, stream);
    scatter_edges<<<(E + 3) / 4, 256, 0, stream>>>(B0, src, dst, dinv, OUT, E);
    finalize_relu<<<NNODES, 256, 0, stream>>>(OUT, B0, dinv, b2);
}